// MaxPooling_40432822124855
// MI455X (gfx1250) — compile-verified
//
#include <hip/hip_runtime.h>
#include <hip/hip_bf16.h>

// ---------------------------------------------------------------------------
// Problem constants (from the reference): VOXEL=4, FULL_SHAPE=(260,346)
// NX=65, NY=87, V=5655, B=8, S=45240, F=64.
// ---------------------------------------------------------------------------
constexpr int NXc = 65;
constexpr int NYc = 87;
constexpr int Vc  = NXc * NYc;     // 5655
constexpr int Bc  = 8;
constexpr int Sc  = Bc * Vc;       // 45240
constexpr int FEAT = 64;

// ---------------------------------------------------------------------------
// TDM (Tensor Data Mover): this toolchain exposes the 6-arg builtin
// (uint32x4 g0, int32x8 g1, int32x4 g2, int32x4 g3, int32x8 g4, i32 cpol).
// We use D# gather mode: 32-bit row indices, up to 8 rows per descriptor.
// ---------------------------------------------------------------------------
#if defined(__has_builtin)
#  if __has_builtin(__builtin_amdgcn_tensor_load_to_lds) && \
      __has_builtin(__builtin_amdgcn_s_wait_tensorcnt)
#    define USE_TDM 1
#  endif
#endif
#ifndef USE_TDM
#  define USE_TDM 0
#endif

typedef unsigned int u32x4 __attribute__((ext_vector_type(4)));
typedef int          i32x8 __attribute__((ext_vector_type(8)));
typedef int          i32x4 __attribute__((ext_vector_type(4)));

__device__ __forceinline__ unsigned lds_byte_offset(const void* p) {
  // LDS aperture: low 32 bits of a flat LDS address are the LDS byte offset.
  return (unsigned)(uintptr_t)p;
}

#if USE_TDM
// Gather-mode D#: pull `nrows` (<=8) rows of FEAT f32 elements from the 2-D
// tensor x[N][FEAT] at 32-bit row indices idx[0..7], into LDS at `lds_off`
// (rows land consecutively: row j -> lds_off + j*FEAT*4).
//   group0: count=1 | gather_index_size=1 (32-bit) | gather_mode=1
//           | lds_addr | global_addr(tensor base) | type=2
//   group1: data_size=4B, tensor_dim0=tile_dim0=FEAT, tensor_dim1=N,
//           tile_dim1=#valid indices, tensor_dim0_stride=FEAT
//   group2/3: row indices 0-3 / 4-7 (32-bit mode)
__device__ __forceinline__ void tdm_gather_rows(unsigned lds_off,
                                                const float* xbase, int N,
                                                const int* idx, int nrows) {
  unsigned long long ga = (unsigned long long)(uintptr_t)xbase;
  u32x4 g0;
  g0[0] = 1u | (1u << 30) | (1u << 31);                  // count | idx32 | gather
  g0[1] = lds_off;                                       // lds_addr
  g0[2] = (unsigned)(ga & 0xFFFFFFFFu);                  // global_addr[31:0]
  g0[3] = (unsigned)((ga >> 32) & 0x1FFFFFFu)            // global_addr[56:32]
          | 0x80000000u;                                 // type=2 ("image")
  unsigned nu = (unsigned)N;
  i32x8 g1;
  g1[0] = 0x20000;                                       // data_size=2 -> 4B
  g1[1] = (int)(((unsigned)FEAT & 0xFFFFu) << 16);       // tensor_dim0 lo16
  g1[2] = (int)((nu & 0xFFFFu) << 16);                   // td0 hi | tensor_dim1 lo
  g1[3] = (int)(((nu >> 16) & 0xFFFFu)                   // tensor_dim1 hi
                | (((unsigned)FEAT & 0xFFFFu) << 16));   // tile_dim0
  g1[4] = (int)((unsigned)nrows & 0xFFFFu);              // tile_dim1 = #indices
  g1[5] = FEAT;                                          // tensor_dim0_stride lo
  g1[6] = 0;
  g1[7] = 0;
  i32x4 g2 = {idx[0], idx[1], idx[2], idx[3]};           // row indices 0-3
  i32x4 g3 = {idx[4], idx[5], idx[6], idx[7]};           // row indices 4-7
  i32x8 gz8 = {0, 0, 0, 0, 0, 0, 0, 0};
  __builtin_amdgcn_tensor_load_to_lds(g0, g1, g2, g3, gz8, 0);
}
#endif

// wave32 butterfly all-reduce sum
__device__ __forceinline__ float waveReduceSum(float v) {
  for (int off = 16; off > 0; off >>= 1) v += __shfl_xor(v, off, 32);
  return v;
}

// ---------------------------------------------------------------------------
// 1) init: zero all bucket counters (points + edges)
// ---------------------------------------------------------------------------
__global__ void init_kernel(unsigned* pcount, unsigned* pfill,
                            unsigned* rcount, unsigned* rfill) {
  int id = blockIdx.x * blockDim.x + threadIdx.x;
  if (id < Sc) { pcount[id] = 0u; pfill[id] = 0u; rcount[id] = 0u; rfill[id] = 0u; }
}

// ---------------------------------------------------------------------------
// 2) cluster ids + point-bucket histogram (1M small atomics, L2-resident)
// ---------------------------------------------------------------------------
__global__ void cluster_kernel(const float* __restrict__ pos,
                               const int* __restrict__ batch,
                               int* __restrict__ cluster,
                               unsigned* pcount, int N) {
  int i = blockIdx.x * blockDim.x + threadIdx.x;
  if (i >= N) return;
  float px = pos[3 * i + 0], py = pos[3 * i + 1];
  int c0 = (int)floorf(px * 0.25f);
  int c1 = (int)floorf(py * 0.25f);
  c0 = min(max(c0, 0), NXc - 1);
  c1 = min(max(c1, 0), NYc - 1);
  int cl = batch[i] * Vc + c0 * NYc + c1;
  cluster[i] = cl;
  atomicAdd(&pcount[cl], 1u);
}

// ---------------------------------------------------------------------------
// 3) generic single-block exclusive scan over S counters (used twice)
// ---------------------------------------------------------------------------
__global__ void scan_kernel(const unsigned* __restrict__ cnt,
                            unsigned* off, int S) {
  __shared__ unsigned sd[1024];
  __shared__ unsigned carry;
  int tid = threadIdx.x;
  if (tid == 0) carry = 0u;
  __syncthreads();
  for (int base = 0; base < S; base += 1024) {
    int i = base + tid;
    unsigned v = (i < S) ? cnt[i] : 0u;
    sd[tid] = v;
    __syncthreads();
    for (int o = 1; o < 1024; o <<= 1) {
      unsigned t = (tid >= o) ? sd[tid - o] : 0u;
      __syncthreads();
      sd[tid] += t;
      __syncthreads();
    }
    unsigned incl = sd[tid];
    unsigned c = carry;
    if (i < S) off[i] = c + incl - v;   // exclusive
    unsigned total = sd[1023];
    __syncthreads();
    if (tid == 0) carry = c + total;
    __syncthreads();
  }
  if (tid == 0) off[S] = carry;
}

// ---------------------------------------------------------------------------
// 4) scatter point indices into their cluster bucket
// ---------------------------------------------------------------------------
__global__ void point_scatter_kernel(const int* __restrict__ cluster,
                                     const unsigned* __restrict__ poffset,
                                     unsigned* pfill, int* pidx, int N) {
  int i = blockIdx.x * blockDim.x + threadIdx.x;
  if (i >= N) return;
  int cl = cluster[i];
  unsigned p = poffset[cl] + atomicAdd(&pfill[cl], 1u);
  pidx[p] = i;
}

// ---------------------------------------------------------------------------
// 5) gather + reduce: one wave32 per cluster. TDM gather-mode DMAs 8 sparse
//    x-rows (256B each) per descriptor into this wave's LDS slice, double-
//    buffered on the wave-private TENSORcnt (no barriers). Lanes cover the
//    64 features; max-reduce in registers; pos/count reduced without any
//    atomics. Single coalesced write per cluster. Also emits batch_out/mask.
//    LDS: 8 waves * 2 buffers * 8 rows * 256B = 32KB.
// ---------------------------------------------------------------------------
__global__ void gather_reduce_kernel(const int* __restrict__ pidx,
                                     const unsigned* __restrict__ poffset,
                                     const float* __restrict__ x,
                                     const float* __restrict__ pos,
                                     float* xpool, float* pospool,
                                     float* batchout, float* maskout, int N) {
  const int WAVES = 8, ROWS = 8;
  __shared__ float buf[WAVES][2][ROWS * FEAT];
  int wave = threadIdx.x >> 5;
  int lane = threadIdx.x & 31;
  int s = blockIdx.x * WAVES + wave;
  if (s >= Sc) return;

  unsigned beg = poffset[s], end = poffset[s + 1];
  int npts = (int)(end - beg);

  float m0 = -3.402823466e38f, m1 = -3.402823466e38f;

#if USE_TDM
  int nbatch = (npts + ROWS - 1) / ROWS;
  if (nbatch > 0) {  // prologue: issue batch 0
    int idx[ROWS];
    int n0 = min(ROWS, npts);
    for (int j = 0; j < ROWS; ++j) idx[j] = (j < n0) ? pidx[beg + j] : 0;
    tdm_gather_rows(lds_byte_offset(&buf[wave][0][0]), x, N, idx, n0);
  }
  for (int b = 0; b < nbatch; ++b) {
    int cur = b & 1;
    if (b + 1 < nbatch) {  // issue next batch into other buffer
      int base = (b + 1) * ROWS;
      int n1 = min(ROWS, npts - base);
      int idx[ROWS];
      for (int j = 0; j < ROWS; ++j) idx[j] = (j < n1) ? pidx[beg + base + j] : 0;
      tdm_gather_rows(lds_byte_offset(&buf[wave][cur ^ 1][0]), x, N, idx, n1);
      __builtin_amdgcn_s_wait_tensorcnt(1);  // batch b complete, b+1 in flight
    } else {
      __builtin_amdgcn_s_wait_tensorcnt(0);
    }
    int nrows = min(ROWS, npts - b * ROWS);
    for (int j = 0; j < nrows; ++j) {
      m0 = fmaxf(m0, buf[wave][cur][j * FEAT + lane]);
      m1 = fmaxf(m1, buf[wave][cur][j * FEAT + lane + 32]);
    }
  }
#else
  for (int i = 0; i < npts; ++i) {  // fallback: direct coalesced row reads
    size_t p = (size_t)pidx[beg + i];
    m0 = fmaxf(m0, x[p * FEAT + lane]);
    m1 = fmaxf(m1, x[p * FEAT + lane + 32]);
  }
#endif

  // pos segment-sum: lanes stride the bucket, butterfly all-reduce
  float sx = 0.0f, sy = 0.0f, sz = 0.0f;
  for (int i = lane; i < npts; i += 32) {
    size_t p = (size_t)pidx[beg + i];
    sx += pos[3 * p + 0];
    sy += pos[3 * p + 1];
    sz += pos[3 * p + 2];
  }
  sx = waveReduceSum(sx);
  sy = waveReduceSum(sy);
  sz = waveReduceSum(sz);

  bool occ = npts > 0;
  xpool[(size_t)s * FEAT + lane]      = occ ? m0 : 0.0f;
  xpool[(size_t)s * FEAT + lane + 32] = occ ? m1 : 0.0f;
  if (lane == 0) {
    float inv = 1.0f / fmaxf((float)npts, 1.0f);
    pospool[3 * s + 0] = sx * inv;
    pospool[3 * s + 1] = sy * inv;
    pospool[3 * s + 2] = sz * inv;
    batchout[s] = (float)(s / Vc);
    maskout[s]  = occ ? 1.0f : 0.0f;
  }
}

// ---------------------------------------------------------------------------
// 6-8) edge coarsening: bucket-by-r histogram, scan (kernel above), scatter
//      of packed keys. key=(r<<16)|c reproduces lexsort((c,r)) order; dup
//      payloads are identical so an unstable within-bucket sort is exact.
// ---------------------------------------------------------------------------
__global__ void edge_count_kernel(const int* __restrict__ ei,
                                  const int* __restrict__ cluster,
                                  unsigned* rcount, int E) {
  int e = blockIdx.x * blockDim.x + threadIdx.x;
  if (e >= E) return;
  atomicAdd(&rcount[cluster[ei[e]]], 1u);
}

__global__ void edge_scatter_kernel(const int* __restrict__ ei,
                                    const int* __restrict__ cluster,
                                    const unsigned* __restrict__ roffset,
                                    unsigned* rfill, unsigned* keys, int E) {
  int e = blockIdx.x * blockDim.x + threadIdx.x;
  if (e >= E) return;
  int r = cluster[ei[e]];
  int c = cluster[ei[E + e]];
  unsigned p = roffset[r] + atomicAdd(&rfill[r], 1u);
  keys[p] = ((unsigned)r << 16) | (unsigned)c;
}

// ---------------------------------------------------------------------------
// 9) one workgroup per r-bucket: LDS bitonic sort + group-local dedupe
//    (valid: equal keys can never straddle an r-bucket boundary).
// ---------------------------------------------------------------------------
__global__ void group_sort_kernel(const unsigned* __restrict__ keys,
                                  const unsigned* __restrict__ roffset,
                                  float* out0, float* out1) {
  __shared__ unsigned lk[8192];
  int r = blockIdx.x;
  unsigned beg = roffset[r], end = roffset[r + 1];
  unsigned lenFull = end - beg;
  if (lenFull == 0) return;
  unsigned len = lenFull > 8192u ? 8192u : lenFull;
  unsigned M = 1;
  while (M < len) M <<= 1;
  for (unsigned i = threadIdx.x; i < M; i += blockDim.x)
    lk[i] = (i < len) ? keys[beg + i] : 0xFFFFFFFFu;
  __syncthreads();
  for (unsigned k = 2; k <= M; k <<= 1) {
    for (unsigned j = k >> 1; j > 0; j >>= 1) {
      for (unsigned i = threadIdx.x; i < M; i += blockDim.x) {
        unsigned ixj = i ^ j;
        if (ixj > i) {
          unsigned a = lk[i], b = lk[ixj];
          bool up = ((i & k) == 0);
          if ((up && a > b) || (!up && a < b)) { lk[i] = b; lk[ixj] = a; }
        }
      }
      __syncthreads();
    }
  }
  for (unsigned i = threadIdx.x; i < len; i += blockDim.x) {
    unsigned key = lk[i];
    unsigned rr = key >> 16, cc = key & 0xFFFFu;
    bool valid = (rr != cc) && (i == 0 ? true : (key != lk[i - 1]));
    unsigned p = beg + i;
    out0[p] = valid ? (float)rr : -1.0f;
    out1[p] = valid ? (float)cc : -1.0f;
  }
  for (unsigned i = threadIdx.x + len; i < lenFull; i += blockDim.x) {
    out0[beg + i] = -1.0f;   // safety valve for >cap buckets
    out1[beg + i] = -1.0f;
  }
}

// ---------------------------------------------------------------------------
// host entry
// ---------------------------------------------------------------------------
extern "C" void kernel_launch(void* const* d_in, const int* in_sizes, int n_in,
                              void* d_out, int out_size, void* d_ws,
                              size_t ws_size, hipStream_t stream) {
  (void)n_in; (void)out_size; (void)ws_size;
  const int N = in_sizes[2];
  const int E = in_sizes[3] / 2;

  const float* x     = (const float*)d_in[0];
  const float* pos   = (const float*)d_in[1];
  const int*   batch = (const int*)d_in[2];
  const int*   ei    = (const int*)d_in[3];

  // output layout: x_pool[S*64] | pos_pool[S*3] | batch_out[S] | edge[2E] | mask[S]
  float* out      = (float*)d_out;
  float* xpool    = out;
  float* pospool  = xpool + (size_t)Sc * FEAT;
  float* batchout = pospool + (size_t)Sc * 3;
  float* edgeout  = batchout + Sc;
  float* out0     = edgeout;
  float* out1     = edgeout + (size_t)E;
  float* maskout  = edgeout + 2 * (size_t)E;

  // workspace layout
  char* w = (char*)d_ws;
  int*      cluster = (int*)w;       w += (size_t)N * 4;
  int*      pidx    = (int*)w;       w += (size_t)N * 4;
  unsigned* pcount  = (unsigned*)w;  w += (size_t)Sc * 4;
  unsigned* pfill   = (unsigned*)w;  w += (size_t)Sc * 4;
  unsigned* poffset = (unsigned*)w;  w += (size_t)(Sc + 1) * 4;
  unsigned* rcount  = (unsigned*)w;  w += (size_t)Sc * 4;
  unsigned* rfill   = (unsigned*)w;  w += (size_t)Sc * 4;
  unsigned* roffset = (unsigned*)w;  w += (size_t)(Sc + 1) * 4;
  w = (char*)(((uintptr_t)w + 255) & ~(uintptr_t)255);
  unsigned* keys    = (unsigned*)w;  // E * 4 bytes

  const int T = 256;
  init_kernel<<<(Sc + T - 1) / T, T, 0, stream>>>(pcount, pfill, rcount, rfill);
  cluster_kernel<<<(N + T - 1) / T, T, 0, stream>>>(pos, batch, cluster,
                                                    pcount, N);
  scan_kernel<<<1, 1024, 0, stream>>>(pcount, poffset, Sc);
  point_scatter_kernel<<<(N + T - 1) / T, T, 0, stream>>>(cluster, poffset,
                                                          pfill, pidx, N);
  gather_reduce_kernel<<<(Sc + 7) / 8, T, 0, stream>>>(pidx, poffset, x, pos,
                                                       xpool, pospool,
                                                       batchout, maskout, N);
  edge_count_kernel<<<(E + T - 1) / T, T, 0, stream>>>(ei, cluster, rcount, E);
  scan_kernel<<<1, 1024, 0, stream>>>(rcount, roffset, Sc);
  edge_scatter_kernel<<<(E + T - 1) / T, T, 0, stream>>>(ei, cluster, roffset,
                                                         rfill, keys, E);
  group_sort_kernel<<<Sc, T, 0, stream>>>(keys, roffset, out0, out1);
}